// ODEBlock_15178414424751
// MI455X (gfx1250) — compile-verified
//
#include <hip/hip_runtime.h>

#define NSTEPS 32
#define LDSW   66   // padded LDS row stride (floats) -> conflict-free column access

typedef float v2f __attribute__((ext_vector_type(2)));
typedef float v8f __attribute__((ext_vector_type(8)));

// alpha*(u) + (1-alpha)*softplus(u), numerically stable softplus.
// Bias is folded in here (u = gemm + c0) so WMMA C can start from inline 0.
__device__ __forceinline__ float smooth_lrelu_bias(float g, float c0) {
    float u  = g + c0;
    float e  = __expf(-fabsf(u));                 // v_exp_f32
    float sp = fmaxf(u, 0.0f) + __logf(1.0f + e); // v_log_f32
    return 0.1f * u + 0.9f * sp;
}

// One ODE-function eval: kout = f(t, <tile currently staged in sYw>).
// A fragments are read from the per-wave LDS scratch (D->A transpose);
// B fragments (W) live permanently in registers; C starts as inline 0.
__device__ __forceinline__ void eval_core(float t, float (&kout)[4][8],
                                          const float* sYw,
                                          const v2f (&Bf)[16][4],
                                          const float (&w0)[4], const float (&bb)[4],
                                          int lo, int hi)
{
    // A fragments (16x4 fp32): lane l -> row m=l&15, K = kc*4 + 2*(l>>4), +1
    v2f A[16];
#pragma unroll
    for (int kc = 0; kc < 16; ++kc)
        A[kc] = *(const v2f*)&sYw[lo * LDSW + kc * 4 + 2 * hi];

    // zero accumulators -> backend uses inline-0 SRC2 on the first WMMA
    v8f acc[4] = {};

    // K=64 in 16 chunks of 4, N=64 in 4 tiles of 16: 64 fp32 WMMAs
#pragma unroll
    for (int kc = 0; kc < 16; ++kc)
#pragma unroll
        for (int nt = 0; nt < 4; ++nt)
            acc[nt] = __builtin_amdgcn_wmma_f32_16x16x4_f32(
                false, A[kc], false, Bf[kc][nt], (short)0, acc[nt], false, false);

    // bias(t)[n] + activation
#pragma unroll
    for (int nt = 0; nt < 4; ++nt) {
        float c0 = fmaf(t, w0[nt], bb[nt]);
#pragma unroll
        for (int v = 0; v < 8; ++v)
            kout[nt][v] = smooth_lrelu_bias(acc[nt][v], c0);
    }
}

// Stage the next eval's input tile into the per-wave LDS scratch.
// D layout: VGPR v, lane l holds (m = v + 8*(l>>4), n = (l&15) + 16*nt).
#define STAGE_TILE(EXPR)                                                \
    do {                                                                \
        _Pragma("unroll")                                               \
        for (int nt = 0; nt < 4; ++nt) {                                \
            _Pragma("unroll")                                           \
            for (int v = 0; v < 8; ++v)                                 \
                sYw[(v + 8 * hi) * LDSW + nt * 16 + lo] = (EXPR);       \
        }                                                               \
    } while (0)

__global__ __launch_bounds__(256)
void odeblock_rk4(const float* __restrict__ x, const float* __restrict__ W,
                  const float* __restrict__ b, float* __restrict__ out)
{
    __shared__ float sWl[64 * LDSW];      // Wl[n][k] = W[n][k+1], padded stride
    __shared__ float sY[8][16 * LDSW];    // per-wave transpose scratch

    const int tid  = threadIdx.x;
    const int wave = tid >> 5;
    const int lane = tid & 31;
    const int lo   = lane & 15;
    const int hi   = lane >> 4;

    float* sYw = sY[wave];

    // cooperative load of W (cols 1..64) into LDS
    for (int i = tid; i < 64 * 64; i += 256) {
        int n = i >> 6, k = i & 63;
        sWl[n * LDSW + k] = W[n * 65 + (k + 1)];
    }
    __syncthreads();

    // hoist ALL B fragments (the whole 64x64 W) into registers: loop-invariant
    // across all 128 evals.  B fragment (4x16): lane l -> n = nt*16 + (l&15),
    // K pair = kc*4 + 2*(l>>4), +1  => one b64 per fragment.
    v2f Bf[16][4];
#pragma unroll
    for (int kc = 0; kc < 16; ++kc)
#pragma unroll
        for (int nt = 0; nt < 4; ++nt)
            Bf[kc][nt] = *(const v2f*)&sWl[(nt * 16 + lo) * LDSW + kc * 4 + 2 * hi];

    // per-lane bias pieces: n = nt*16 + lo
    float w0[4], bb[4];
#pragma unroll
    for (int nt = 0; nt < 4; ++nt) {
        int n  = nt * 16 + lo;
        w0[nt] = W[n * 65];   // t-column
        bb[nt] = b[n];
    }

    const int rowBase = blockIdx.x * 128 + wave * 16;

    // load y0 directly into D-layout registers (coalesced 64B half-wave reads)
    float y[4][8];
#pragma unroll
    for (int nt = 0; nt < 4; ++nt)
#pragma unroll
        for (int v = 0; v < 8; ++v) {
            int m = v + 8 * hi;
            y[nt][v] = x[(size_t)(rowBase + m) * 64 + nt * 16 + lo];
        }

    float kst[4][8], ksum[4][8];
    const float h = 1.0f / (float)NSTEPS;

    for (int s = 0; s < NSTEPS; ++s) {
        float t0 = (float)s * h;

        // ---- k1 : input = y
        STAGE_TILE(y[nt][v]);
        eval_core(t0, kst, sYw, Bf, w0, bb, lo, hi);
#pragma unroll
        for (int nt = 0; nt < 4; ++nt)
#pragma unroll
            for (int v = 0; v < 8; ++v) ksum[nt][v] = kst[nt][v];

        // ---- k2 : input = y + h/2 * k1 (computed on the fly into LDS)
        STAGE_TILE(fmaf(0.5f * h, kst[nt][v], y[nt][v]));
        eval_core(t0 + 0.5f * h, kst, sYw, Bf, w0, bb, lo, hi);
#pragma unroll
        for (int nt = 0; nt < 4; ++nt)
#pragma unroll
            for (int v = 0; v < 8; ++v)
                ksum[nt][v] = fmaf(2.0f, kst[nt][v], ksum[nt][v]);

        // ---- k3 : input = y + h/2 * k2
        STAGE_TILE(fmaf(0.5f * h, kst[nt][v], y[nt][v]));
        eval_core(t0 + 0.5f * h, kst, sYw, Bf, w0, bb, lo, hi);
#pragma unroll
        for (int nt = 0; nt < 4; ++nt)
#pragma unroll
            for (int v = 0; v < 8; ++v)
                ksum[nt][v] = fmaf(2.0f, kst[nt][v], ksum[nt][v]);

        // ---- k4 : input = y + h * k3
        STAGE_TILE(fmaf(h, kst[nt][v], y[nt][v]));
        eval_core(t0 + h, kst, sYw, Bf, w0, bb, lo, hi);
#pragma unroll
        for (int nt = 0; nt < 4; ++nt)
#pragma unroll
            for (int v = 0; v < 8; ++v) {
                ksum[nt][v] += kst[nt][v];
                y[nt][v]    = fmaf(h * (1.0f / 6.0f), ksum[nt][v], y[nt][v]);
            }
    }

    // final state -> global (coalesced)
#pragma unroll
    for (int nt = 0; nt < 4; ++nt)
#pragma unroll
        for (int v = 0; v < 8; ++v) {
            int m = v + 8 * hi;
            out[(size_t)(rowBase + m) * 64 + nt * 16 + lo] = y[nt][v];
        }
}

extern "C" void kernel_launch(void* const* d_in, const int* in_sizes, int n_in,
                              void* d_out, int out_size, void* d_ws, size_t ws_size,
                              hipStream_t stream) {
    (void)n_in; (void)out_size; (void)d_ws; (void)ws_size;
    const float* x = (const float*)d_in[0];   // [B, 64]
    const float* W = (const float*)d_in[1];   // [64, 65]
    const float* b = (const float*)d_in[2];   // [64]
    float* out     = (float*)d_out;           // [B, 64]

    const int batch  = in_sizes[0] / 64;      // 262144
    const int blocks = batch / 128;           // 128 rows per block (8 waves x 16)
    odeblock_rk4<<<blocks, 256, 0, stream>>>(x, W, b, out);
}